// SpectralConv3d_66580583022979
// MI455X (gfx1250) — compile-verified
//
#include <hip/hip_runtime.h>

typedef __attribute__((ext_vector_type(2))) float v2f;
typedef __attribute__((ext_vector_type(8))) float v8f;

#define PI_F 3.14159265358979323846f
// Problem constants: B=4, D1=D2=D3=64, CIN=COUT=32, M1=M2=M3=8
// 32 "slices" = (b, a1) with b<4, a1<8 (only d1<8 contributes to output).

__device__ __forceinline__ size_t slice_out_base(int slice) {
    int b = slice >> 3, a1 = slice & 7;
    return (size_t)(b * 64 + a1) * 131072;   // y[b][a1][0][0][0]
}

// ---------------------------------------------------------------- zero output
__global__ void k_zero(float4* __restrict__ out, int n4) {
    int i = blockIdx.x * blockDim.x + threadIdx.x;
    int stride = gridDim.x * blockDim.x;
    float4 z = {0.f, 0.f, 0.f, 0.f};
    for (; i < n4; i += stride) out[i] = z;
}

// ------------------------------------------- Wsum[c][o][k] = sum_{i,j} w[c,o,i,j,k]
__global__ void k_wsum(const float* __restrict__ wr, const float* __restrict__ wi,
                       float* __restrict__ Wr, float* __restrict__ Wi) {
    int t = blockIdx.x * blockDim.x + threadIdx.x;   // 8192 = 32*32*8
    if (t >= 32 * 32 * 8) return;
    int k = t & 7;
    int o = (t >> 3) & 31;
    int c = t >> 8;
    const int base = (c * 32 + o) * 512 + k;         // w[(c,o,i,j,k)] = (c*32+o)*512 + (i*8+j)*8 + k
    float sr = 0.f, si = 0.f;
    for (int ij = 0; ij < 64; ++ij) {
        sr += wr[base + ij * 8];
        si += wi[base + ij * 8];
    }
    Wr[(c * 32 + o) * 8 + k] = sr;
    Wi[(c * 32 + o) * 8 + k] = si;
}

// ---- M_q3[oc][c] = sum_{o,qc} exp(+2pi i (o*oc - qc*c)/32) * Wsum[qc][o][q3]
__global__ void __launch_bounds__(256) k_mmat(const float* __restrict__ Wr,
                                              const float* __restrict__ Wi,
                                              float* __restrict__ Mr,
                                              float* __restrict__ Mi) {
    __shared__ float c32[32], s32[32];
    int t = threadIdx.x;
    if (t < 32) {
        float a = 2.f * PI_F * (float)t / 32.f;
        c32[t] = cosf(a);
        s32[t] = sinf(a);
    }
    __syncthreads();
    int gid = blockIdx.x * blockDim.x + t;           // 8192 = 8*32*32
    int c  = gid & 31;
    int oc = (gid >> 5) & 31;
    int q3 = gid >> 10;
    float ar = 0.f, ai = 0.f;
    for (int o = 0; o < 32; ++o) {
        for (int qc = 0; qc < 32; ++qc) {
            int n = (o * oc - qc * c) & 31;          // mod-32 (two's complement safe)
            float tr = c32[n], ti = s32[n];
            float wrv = Wr[(qc * 32 + o) * 8 + q3];
            float wiv = Wi[(qc * 32 + o) * 8 + q3];
            ar += tr * wrv - ti * wiv;
            ai += tr * wiv + ti * wrv;
        }
    }
    Mr[q3 * 1024 + oc * 32 + c] = ar;
    Mi[q3 * 1024 + oc * 32 + c] = ai;
}

// ---------------- forward: one block per slice; x[64][64][32] -> Xhat[q2<8][q3<8][c]
__global__ void __launch_bounds__(256) k_fwd(const float* __restrict__ x,
                                             float* __restrict__ Xr,
                                             float* __restrict__ Xi) {
    extern __shared__ float lds[];
    float* c64 = lds;                 // 64
    float* s64 = lds + 64;            // 64
    float* Ar  = lds + 128;           // 64*8*32 = 16384
    float* Ai  = Ar + 16384;          // 16384   (total ~129 KB, fits 320 KB LDS)
    int t = threadIdx.x;
    int slice = blockIdx.x;           // b*8 + a1
    if (t < 64) {
        float a = 2.f * PI_F * (float)t / 64.f;
        c64[t] = cosf(a);
        s64[t] = sinf(a);
    }
    __syncthreads();
    int b  = slice >> 3;
    int a1 = slice & 7;
    const float* xs = x + (size_t)(b * 64 + a1) * 131072;   // x[b][a1][.][.][.]

    // Stage A: A[d2][q3][c] = sum_d3 x[d2][d3][c] * e^{-2pi i q3 d3/64}
    for (int p8 = 0; p8 < 8; ++p8) {
        int p  = t + 256 * p8;        // 2048 (d2,c) pairs
        int d2 = p >> 5;
        int c  = p & 31;
        float ar[8], ai[8];
        #pragma unroll
        for (int q = 0; q < 8; ++q) { ar[q] = 0.f; ai[q] = 0.f; }
        const float* row = xs + d2 * 2048 + c;
        for (int d3 = 0; d3 < 64; ++d3) {
            float v = row[d3 * 32];
            #pragma unroll
            for (int q = 0; q < 8; ++q) {
                int n = (q * d3) & 63;
                ar[q] += v * c64[n];
                ai[q] -= v * s64[n];
            }
        }
        #pragma unroll
        for (int q = 0; q < 8; ++q) {
            Ar[(d2 * 8 + q) * 32 + c] = ar[q];
            Ai[(d2 * 8 + q) * 32 + c] = ai[q];
        }
    }
    __syncthreads();

    // Stage B: X[q2][q3][c] = sum_d2 A[d2][q3][c] * e^{-2pi i q2 d2/64}
    for (int j = 0; j < 8; ++j) {
        int o  = t + 256 * j;         // 2048 outputs
        int c  = o & 31;
        int q3 = (o >> 5) & 7;
        int q2 = o >> 8;
        float br = 0.f, bi = 0.f;
        for (int d2 = 0; d2 < 64; ++d2) {
            int n = (q2 * d2) & 63;
            float arv = Ar[(d2 * 8 + q3) * 32 + c];
            float aiv = Ai[(d2 * 8 + q3) * 32 + c];
            br += arv * c64[n] + aiv * s64[n];
            bi += aiv * c64[n] - arv * s64[n];
        }
        Xr[slice * 2048 + o] = br;
        Xi[slice * 2048 + o] = bi;
    }
}

// ------------- channel mixing via fp32 WMMA: per q3, C(256x32) = X(256x32) x M_q3^T
// complex: Cr = Xr*Mr^T - Xi*Mi^T ; Ci = Xr*Mi^T + Xi*Mr^T
// site s = slice*8 + q2 (0..255); X stored at [s*256 + q3*32 + c]
__global__ void __launch_bounds__(32) k_mix(const float* __restrict__ Xr,
                                            const float* __restrict__ Xi,
                                            const float* __restrict__ Mr,
                                            const float* __restrict__ Mi,
                                            float* __restrict__ Cr,
                                            float* __restrict__ Ci) {
    int bid   = blockIdx.x;           // 256 waves: 8 q3 * 16 mtiles * 2 ntiles
    int q3    = bid >> 5;
    int mtile = (bid >> 1) & 15;
    int ntile = bid & 1;
    int l      = threadIdx.x;         // wave32 lane
    int half   = l >> 4;
    int lane15 = l & 15;
    int row = mtile * 16 + lane15;    // A: M index (site)
    int n   = ntile * 16 + lane15;    // B: N index (oc)
    int ko  = half * 2;               // K split across lane halves
    const float* xr = Xr + row * 256 + q3 * 32;
    const float* xi = Xi + row * 256 + q3 * 32;
    const float* mr = Mr + q3 * 1024 + n * 32;   // M_q3[oc=n][c]
    const float* mi = Mi + q3 * 1024 + n * 32;
    v8f cr = {};
    v8f ci = {};
    #pragma unroll
    for (int ks = 0; ks < 8; ++ks) {
        int kk = ks * 4 + ko;
        v2f aR = { xr[kk], xr[kk + 1] };
        v2f aI = { xi[kk], xi[kk + 1] };
        v2f bR = { mr[kk], mr[kk + 1] };
        v2f bI = { mi[kk], mi[kk + 1] };
        v2f nbI = { -bI.x, -bI.y };
        cr = __builtin_amdgcn_wmma_f32_16x16x4_f32(false, aR, false, bR,  (short)0, cr, false, false);
        cr = __builtin_amdgcn_wmma_f32_16x16x4_f32(false, aI, false, nbI, (short)0, cr, false, false);
        ci = __builtin_amdgcn_wmma_f32_16x16x4_f32(false, aR, false, bI,  (short)0, ci, false, false);
        ci = __builtin_amdgcn_wmma_f32_16x16x4_f32(false, aI, false, bR,  (short)0, ci, false, false);
    }
    #pragma unroll
    for (int r = 0; r < 8; ++r) {
        int m = r + half * 8;                     // D: vgpr r holds rows r (lanes 0-15) / r+8 (16-31)
        int s = mtile * 16 + m;
        Cr[s * 256 + q3 * 32 + n] = cr[r];
        Ci[s * 256 + q3 * 32 + n] = ci[r];
    }
}

// ---------------- inverse: one block per slice; C[q2][q3][oc] -> y[d2][d3][oc] (real)
__global__ void __launch_bounds__(256) k_inv(const float* __restrict__ Cr,
                                             const float* __restrict__ Ci,
                                             float* __restrict__ y) {
    extern __shared__ float lds[];
    float* c64 = lds;                 // 64
    float* s64 = lds + 64;            // 64
    float* CrL = lds + 128;           // 2048
    float* CiL = CrL + 2048;          // 2048
    float* Er  = CiL + 2048;          // 8*64*32 = 16384
    float* Ei  = Er + 16384;          // 16384   (total ~148 KB)
    int t = threadIdx.x;
    int slice = blockIdx.x;
    if (t < 64) {
        float a = 2.f * PI_F * (float)t / 64.f;
        c64[t] = cosf(a);
        s64[t] = sinf(a);
    }
    for (int j = 0; j < 8; ++j) {
        int o = t + 256 * j;
        CrL[o] = Cr[slice * 2048 + o];
        CiL[o] = Ci[slice * 2048 + o];
    }
    __syncthreads();

    // Stage A: E[q2][d3][oc] = sum_{q3<8} C[q2][q3][oc] * e^{+2pi i q3 d3/64}
    for (int j = 0; j < 64; ++j) {
        int idx = t + 256 * j;        // 16384 outputs
        int oc = idx & 31;
        int d3 = (idx >> 5) & 63;
        int q2 = idx >> 11;
        float er = 0.f, ei = 0.f;
        #pragma unroll
        for (int q3 = 0; q3 < 8; ++q3) {
            int n = (q3 * d3) & 63;
            float crv = CrL[q2 * 256 + q3 * 32 + oc];
            float civ = CiL[q2 * 256 + q3 * 32 + oc];
            er += crv * c64[n] - civ * s64[n];
            ei += crv * s64[n] + civ * c64[n];
        }
        Er[idx] = er;
        Ei[idx] = ei;
    }
    __syncthreads();

    // Stage B: y[d2][d3][oc] = (1/131072) * Re( sum_{q2<8} E[q2][d3][oc] * e^{+2pi i q2 d2/64} )
    float* ys = y + slice_out_base(slice);
    for (int j = 0; j < 512; ++j) {
        int idx = t + 256 * j;        // 131072 outputs
        int oc = idx & 31;
        int d3 = (idx >> 5) & 63;
        int d2 = idx >> 11;
        float acc = 0.f;
        #pragma unroll
        for (int q2 = 0; q2 < 8; ++q2) {
            int n = (q2 * d2) & 63;
            acc += Er[(q2 * 64 + d3) * 32 + oc] * c64[n]
                 - Ei[(q2 * 64 + d3) * 32 + oc] * s64[n];
        }
        ys[idx] = acc * (1.f / 131072.f);
    }
}

extern "C" void kernel_launch(void* const* d_in, const int* in_sizes, int n_in,
                              void* d_out, int out_size, void* d_ws, size_t ws_size,
                              hipStream_t stream) {
    (void)in_sizes; (void)n_in; (void)ws_size;
    const float* x  = (const float*)d_in[0];
    const float* wr = (const float*)d_in[1];
    const float* wi = (const float*)d_in[2];
    float* out = (float*)d_out;

    float* ws = (float*)d_ws;
    float* Wr = ws;              // 8192
    float* Wi = Wr + 8192;       // 8192
    float* Mr = Wi + 8192;       // 8192
    float* Mi = Mr + 8192;       // 8192
    float* Xr = Mi + 8192;       // 65536
    float* Xi = Xr + 65536;      // 65536
    float* Cr = Xi + 65536;      // 65536
    float* Ci = Cr + 65536;      // 65536  -> ~1.2 MB total

    size_t lds_fwd = (size_t)(128 + 2 * 16384) * sizeof(float);               // ~129 KB
    size_t lds_inv = (size_t)(128 + 2 * 2048 + 2 * 16384) * sizeof(float);    // ~148 KB
    (void)hipFuncSetAttribute((const void*)k_fwd, hipFuncAttributeMaxDynamicSharedMemorySize, (int)lds_fwd);
    (void)hipFuncSetAttribute((const void*)k_inv, hipFuncAttributeMaxDynamicSharedMemorySize, (int)lds_inv);

    k_zero<<<2048, 256, 0, stream>>>((float4*)out, out_size / 4);
    k_wsum<<<32, 256, 0, stream>>>(wr, wi, Wr, Wi);
    k_mmat<<<32, 256, 0, stream>>>(Wr, Wi, Mr, Mi);
    k_fwd<<<32, 256, lds_fwd, stream>>>(x, Xr, Xi);
    k_mix<<<256, 32, 0, stream>>>(Xr, Xi, Mr, Mi, Cr, Ci);
    k_inv<<<32, 256, lds_inv, stream>>>(Cr, Ci, out);
}